// LSTMTagger_65481071395876
// MI455X (gfx1250) — compile-verified
//
#include <hip/hip_runtime.h>
#include <math.h>

typedef __attribute__((ext_vector_type(2))) float v2f;
typedef __attribute__((ext_vector_type(8))) float v8f;

#define S_LEN 8192
#define T_TAGS 8192
#define XSTR 6

__device__ __forceinline__ float fast_sigmoid(float x) {
    return 1.0f / (1.0f + __expf(-x));
}
__device__ __forceinline__ float fast_tanh(float x) {
    return 2.0f / (1.0f + __expf(-2.0f * x)) - 1.0f;
}

// ---------------------------------------------------------------------------
// Kernel 0: attn_scalar — 4-qubit statevector sim (16 complex amps), 1 lane.
// attn = 1024 * sum_idx |amp[idx]|^2 * popcount(idx)   (P1 rows sum to popcount)
// ---------------------------------------------------------------------------
__global__ void attn_kernel(const float* __restrict__ rotation,
                            const float* __restrict__ entangle,
                            float* __restrict__ attn_out) {
    if (threadIdx.x != 0 || blockIdx.x != 0) return;
    float re[16], im[16];
    #pragma unroll
    for (int i = 0; i < 16; i++) { re[i] = 0.0f; im[i] = 0.0f; }
    re[0] = 1.0f;
    for (int q = 0; q < 4; q++) {
        float rx = rotation[3*q], ry = rotation[3*q+1], rz = rotation[3*q+2];
        int stride = 1 << (3 - q);
        // RX: [[c, -i s],[-i s, c]]
        float c = __cosf(0.5f*rx), s = __sinf(0.5f*rx);
        for (int p = 0; p < 16; p++) if (!(p & stride)) {
            int pq = p | stride;
            float ar=re[p], ai=im[p], br=re[pq], bi=im[pq];
            re[p]  =  c*ar + s*bi;  im[p]  =  c*ai - s*br;
            re[pq] =  s*ai + c*br;  im[pq] = -s*ar + c*bi;
        }
        // RY: [[c, -s],[s, c]]
        c = __cosf(0.5f*ry); s = __sinf(0.5f*ry);
        for (int p = 0; p < 16; p++) if (!(p & stride)) {
            int pq = p | stride;
            float ar=re[p], ai=im[p], br=re[pq], bi=im[pq];
            re[p]  = c*ar - s*br;  im[p]  = c*ai - s*bi;
            re[pq] = s*ar + c*br;  im[pq] = s*ai + c*bi;
        }
        // RZ: diag(e^{-i rz/2}, e^{+i rz/2})
        float er = __cosf(0.5f*rz), ei = __sinf(0.5f*rz);
        for (int p = 0; p < 16; p++) if (!(p & stride)) {
            int pq = p | stride;
            float ar=re[p], ai=im[p], br=re[pq], bi=im[pq];
            re[p]  = er*ar + ei*ai;  im[p]  = er*ai - ei*ar;
            re[pq] = er*br - ei*bi;  im[pq] = er*bi + ei*br;
        }
    }
    // CRX chain: control q (bit 3-q), target q+1 (bit 2-q); RX block on control==1
    for (int q = 0; q < 3; q++) {
        float th = entangle[q];
        float c = __cosf(0.5f*th), s = __sinf(0.5f*th);
        int cb = 1 << (3 - q), tb = 1 << (2 - q);
        for (int p = 0; p < 16; p++) if ((p & cb) && !(p & tb)) {
            int pq = p | tb;
            float ar=re[p], ai=im[p], br=re[pq], bi=im[pq];
            re[p]  =  c*ar + s*bi;  im[p]  =  c*ai - s*br;
            re[pq] =  s*ai + c*br;  im[pq] = -s*ar + c*bi;
        }
    }
    float acc = 0.0f;
    for (int p = 0; p < 16; p++)
        acc += (re[p]*re[p] + im[p]*im[p]) * (float)__popc(p);
    attn_out[0] = 1024.0f * acc;
}

// ---------------------------------------------------------------------------
// Kernel 1: build x[t][6] = {emb[sentence[t]][0..3], conv, attn}
// ---------------------------------------------------------------------------
__global__ void build_x_kernel(const int* __restrict__ sentence,
                               const float* __restrict__ emb,
                               const float* __restrict__ attn_ptr,
                               float* __restrict__ x) {
    int t = blockIdx.x * blockDim.x + threadIdx.x;
    if (t >= S_LEN) return;
    int idx = sentence[t];
    float e0 = emb[idx*4+0], e1 = emb[idx*4+1], e2 = emb[idx*4+2], e3 = emb[idx*4+3];
    float conv = 0.25f * (float)((e0 > 0.5f) + (e1 > 0.5f) + (e2 > 0.5f) + (e3 > 0.5f));
    float at = attn_ptr[0];
    float* xr = x + (size_t)t * XSTR;
    xr[0]=e0; xr[1]=e1; xr[2]=e2; xr[3]=e3; xr[4]=conv; xr[5]=at;
}

// ---------------------------------------------------------------------------
// Kernel 2: LSTM scan, one wave32. Closed-form qlayer:
//   c_w = cos( Wg[g][w]·comb + bg[g][w] + thg[g][w] )
//   meas[g] = [c1c2c3, c0c1, c0c1c2, c0c1c2c3]  (CNOT-ring Heisenberg result)
// Lane l computes (g,w) = (l>>2, l&3) cosine (lanes 16..31 mirror 0..15);
// lane group (l&3) carries wire state hx/cx redundantly -> zero divergence.
// ---------------------------------------------------------------------------
__global__ __launch_bounds__(32) void scan_kernel(const float* __restrict__ Wg,
                                                  const float* __restrict__ bg,
                                                  const float* __restrict__ thg,
                                                  const float* __restrict__ x,
                                                  float* __restrict__ outs) {
    int lane = threadIdx.x;
    int gw = lane & 15;           // (g*4 + w)
    int wq = lane & 3;            // wire index this lane tracks
    float w0 = Wg[gw*10+0], w1 = Wg[gw*10+1], w2 = Wg[gw*10+2], w3 = Wg[gw*10+3];
    float w4 = Wg[gw*10+4], w5 = Wg[gw*10+5], w6 = Wg[gw*10+6], w7 = Wg[gw*10+7];
    float w8 = Wg[gw*10+8], w9 = Wg[gw*10+9];
    float bias = bg[gw] + thg[gw];
    float hx = 0.0f, cx = 0.0f;

    for (int t = 0; t < S_LEN; t++) {
        const float* xr = x + (size_t)t * XSTR;
        float x0 = xr[0], x1 = xr[1], x2 = xr[2], x3 = xr[3], x4 = xr[4], x5 = xr[5];
        float h0 = __shfl(hx, 0, 32), h1 = __shfl(hx, 1, 32);
        float h2 = __shfl(hx, 2, 32), h3 = __shfl(hx, 3, 32);
        float tot = bias + w0*x0 + w1*x1 + w2*x2 + w3*x3 + w4*x4 + w5*x5
                         + w6*h0 + w7*h1 + w8*h2 + w9*h3;
        float cth = __cosf(tot);
        float cc[16];
        #pragma unroll
        for (int k = 0; k < 16; k++) cc[k] = __shfl(cth, k, 32);

        float m[4];
        #pragma unroll
        for (int g = 0; g < 4; g++) {
            float c0 = cc[4*g+0], c1 = cc[4*g+1], c2 = cc[4*g+2], c3 = cc[4*g+3];
            float p01 = c0 * c1;
            float m0 = c1 * c2 * c3;     // <Z0> = Z1 Z2 Z3
            float m1 = p01;              // <Z1> = Z0 Z1
            float m2 = p01 * c2;         // <Z2> = Z0 Z1 Z2
            float m3 = m2 * c3;          // <Z3> = Z0 Z1 Z2 Z3
            m[g] = (wq == 0) ? m0 : (wq == 1) ? m1 : (wq == 2) ? m2 : m3;
        }
        float fg = fast_sigmoid(m[0]);
        float ig = fast_sigmoid(m[1]);
        float gg = fast_tanh(m[2]);
        float og = fast_sigmoid(m[3]);
        cx = fg * cx + ig * gg;
        hx = og * fast_tanh(cx);
        if (lane < 4) outs[(size_t)t * 4 + lane] = hx;
    }
}

// ---------------------------------------------------------------------------
// Kernel 3: logits = outs(8192x4) @ W_tag^T(4x8192) + b_tag, log_softmax rows.
// V_WMMA_F32_16X16X4_F32, one block = 16 rows x all 8192 tags, 8 waves split N.
// 3 passes (max / sumexp / write); logits recomputed (4 FMA/elt via WMMA ~ free),
// only the 256 MB final write streams to HBM.
// ---------------------------------------------------------------------------
__global__ __launch_bounds__(256) void tag_softmax_kernel(const float* __restrict__ outs,
                                                          const float* __restrict__ W_tag,
                                                          const float* __restrict__ b_tag,
                                                          float* __restrict__ out) {
    __shared__ float pm[16][8];
    __shared__ float ps[16][8];
    __shared__ float rowM[16];
    __shared__ float rowLZ[16];

    const int lane = threadIdx.x & 31;
    const int wv   = threadIdx.x >> 5;
    const int half = lane >> 4;          // 0: K=0,1 rows 0..7 ; 1: K=2,3 rows 8..15
    const int l15  = lane & 15;
    const int k0   = half * 2;
    const int rowBase = blockIdx.x * 16;
    const int NT = T_TAGS / 16;          // 512 column tiles

    // A tile (16x4 f32): lane m -> K=k0,k0+1 of row m
    v2f a;
    a.x = outs[(size_t)(rowBase + l15) * 4 + k0];
    a.y = outs[(size_t)(rowBase + l15) * 4 + k0 + 1];

    // ---------------- pass 1: row max ----------------
    float mloc[8];
    #pragma unroll
    for (int r = 0; r < 8; r++) mloc[r] = -3.0e38f;
    for (int tile = wv; tile < NT; tile += 8) {
        int n = tile * 16 + l15;
        v2f b; b.x = W_tag[(size_t)n * 4 + k0]; b.y = W_tag[(size_t)n * 4 + k0 + 1];
        float bt = b_tag[n];
        v8f c = {bt, bt, bt, bt, bt, bt, bt, bt};
        c = __builtin_amdgcn_wmma_f32_16x16x4_f32(false, a, false, b, (short)0, c, false, false);
        const float* cp = (const float*)&c;
        #pragma unroll
        for (int r = 0; r < 8; r++) mloc[r] = fmaxf(mloc[r], cp[r]);
    }
    #pragma unroll
    for (int off = 1; off < 16; off <<= 1)
        #pragma unroll
        for (int r = 0; r < 8; r++) mloc[r] = fmaxf(mloc[r], __shfl_xor(mloc[r], off, 32));
    if (l15 == 0) {
        #pragma unroll
        for (int r = 0; r < 8; r++) pm[r + half * 8][wv] = mloc[r];
    }
    __syncthreads();
    if (threadIdx.x < 16) {
        float m = pm[threadIdx.x][0];
        #pragma unroll
        for (int w = 1; w < 8; w++) m = fmaxf(m, pm[threadIdx.x][w]);
        rowM[threadIdx.x] = m;
    }
    __syncthreads();

    // ---------------- pass 2: sum exp ----------------
    float Mr[8];
    #pragma unroll
    for (int r = 0; r < 8; r++) Mr[r] = rowM[r + half * 8];
    float sloc[8];
    #pragma unroll
    for (int r = 0; r < 8; r++) sloc[r] = 0.0f;
    for (int tile = wv; tile < NT; tile += 8) {
        int n = tile * 16 + l15;
        v2f b; b.x = W_tag[(size_t)n * 4 + k0]; b.y = W_tag[(size_t)n * 4 + k0 + 1];
        float bt = b_tag[n];
        v8f c = {bt, bt, bt, bt, bt, bt, bt, bt};
        c = __builtin_amdgcn_wmma_f32_16x16x4_f32(false, a, false, b, (short)0, c, false, false);
        const float* cp = (const float*)&c;
        #pragma unroll
        for (int r = 0; r < 8; r++) sloc[r] += __expf(cp[r] - Mr[r]);
    }
    #pragma unroll
    for (int off = 1; off < 16; off <<= 1)
        #pragma unroll
        for (int r = 0; r < 8; r++) sloc[r] += __shfl_xor(sloc[r], off, 32);
    if (l15 == 0) {
        #pragma unroll
        for (int r = 0; r < 8; r++) ps[r + half * 8][wv] = sloc[r];
    }
    __syncthreads();
    if (threadIdx.x < 16) {
        float s = 0.0f;
        #pragma unroll
        for (int w = 0; w < 8; w++) s += ps[threadIdx.x][w];
        rowLZ[threadIdx.x] = rowM[threadIdx.x] + __logf(s);
    }
    __syncthreads();

    // ---------------- pass 3: write logit - logZ ----------------
    float LZ[8];
    #pragma unroll
    for (int r = 0; r < 8; r++) LZ[r] = rowLZ[r + half * 8];
    for (int tile = wv; tile < NT; tile += 8) {
        int n = tile * 16 + l15;
        v2f b; b.x = W_tag[(size_t)n * 4 + k0]; b.y = W_tag[(size_t)n * 4 + k0 + 1];
        float bt = b_tag[n];
        v8f c = {bt, bt, bt, bt, bt, bt, bt, bt};
        c = __builtin_amdgcn_wmma_f32_16x16x4_f32(false, a, false, b, (short)0, c, false, false);
        const float* cp = (const float*)&c;
        #pragma unroll
        for (int r = 0; r < 8; r++) {
            int row = r + half * 8;
            out[(size_t)(rowBase + row) * T_TAGS + n] = cp[r] - LZ[r];
        }
    }
}

// ---------------------------------------------------------------------------
extern "C" void kernel_launch(void* const* d_in, const int* in_sizes, int n_in,
                              void* d_out, int out_size, void* d_ws, size_t ws_size,
                              hipStream_t stream) {
    const int*   sentence = (const int*)d_in[0];
    const float* emb      = (const float*)d_in[1];
    const float* Wg       = (const float*)d_in[2];
    const float* bg       = (const float*)d_in[3];
    const float* thg      = (const float*)d_in[4];
    const float* rotation = (const float*)d_in[5];
    const float* entangle = (const float*)d_in[6];
    const float* W_tag    = (const float*)d_in[7];
    const float* b_tag    = (const float*)d_in[8];
    float* out = (float*)d_out;

    char* ws = (char*)d_ws;
    float* attn = (float*)ws;                                            // 1 float
    float* x    = (float*)(ws + 256);                                    // 8192*6 f32
    float* outs = (float*)(ws + 256 + (size_t)S_LEN * XSTR * sizeof(float)); // 8192*4 f32

    attn_kernel<<<1, 32, 0, stream>>>(rotation, entangle, attn);
    build_x_kernel<<<(S_LEN + 255) / 256, 256, 0, stream>>>(sentence, emb, attn, x);
    scan_kernel<<<1, 32, 0, stream>>>(Wg, bg, thg, x, outs);
    tag_softmax_kernel<<<S_LEN / 16, 256, 0, stream>>>(outs, W_tag, b_tag, out);
}